// Refiner_7576322310451
// MI455X (gfx1250) — compile-verified
//
#include <hip/hip_runtime.h>
#include <hip/hip_bf16.h>
#include <math.h>

// Planck*c in J*m, as float32 (reference computes in f32 after weak-typing)
#define HC_F 1.98644586e-25f

typedef float v2f __attribute__((ext_vector_type(2)));
typedef float v4f __attribute__((ext_vector_type(4)));
typedef float v8f __attribute__((ext_vector_type(8)));

#define WAVES_PER_BLOCK 8
#define CHUNK_PAIRS 16          // 16 tile-pairs = 32 column tiles per wave chunk
#define ROW_TILES_PER_WAVE 4

// Bare single-instruction maxes (finite data -> no sNaN canonicalization needed).
__device__ __forceinline__ float fmax_raw(float a, float b) {
    float r;
    asm("v_max_num_f32 %0, %1, %2" : "=v"(r) : "v"(a), "v"(b));
    return r;
}
__device__ __forceinline__ float fmax3_raw(float a, float b, float c) {
    float r;
    asm("v_max3_num_f32 %0, %1, %2, %3" : "=v"(r) : "v"(a), "v"(b), "v"(c));
    return r;
}

// ---------------------------------------------------------------------------
// Kernel 0: reciprocal lattice in lab frame (3x3), single thread.
// RL = Rx(t0)Ry(t1)Rz(t2) @ inv(prim)^T, row-major into ws[0..8].
// ---------------------------------------------------------------------------
__device__ __forceinline__ void mat3mul(float o[3][3], const float a[3][3], const float b[3][3]) {
    #pragma unroll
    for (int i = 0; i < 3; ++i)
        #pragma unroll
        for (int j = 0; j < 3; ++j)
            o[i][j] = a[i][0] * b[0][j] + a[i][1] * b[1][j] + a[i][2] * b[2][j];
}

__global__ void prep_rl_kernel(const float* __restrict__ lat, const float* __restrict__ ang,
                               float* __restrict__ rl) {
    if (threadIdx.x != 0 || blockIdx.x != 0) return;
    float a = lat[0], b = lat[1], c = lat[2];
    float cal = cosf(lat[3]), cbe = cosf(lat[4]);
    float cga = cosf(lat[5]), sga = sinf(lat[5]);
    float e3y = (cal - cbe * cga) / sga;
    float e3z = sqrtf(fmaxf(1.0f - cbe * cbe - e3y * e3y, 1e-12f));
    float P[3][3] = {{a, b * cga, c * cbe},
                     {0.0f, b * sga, c * e3y},
                     {0.0f, 0.0f, c * e3z}};
    float det = P[0][0] * (P[1][1] * P[2][2] - P[1][2] * P[2][1])
              - P[0][1] * (P[1][0] * P[2][2] - P[1][2] * P[2][0])
              + P[0][2] * (P[1][0] * P[2][1] - P[1][1] * P[2][0]);
    float id = 1.0f / det;
    float inv[3][3];
    inv[0][0] =  (P[1][1] * P[2][2] - P[1][2] * P[2][1]) * id;
    inv[0][1] = -(P[0][1] * P[2][2] - P[0][2] * P[2][1]) * id;
    inv[0][2] =  (P[0][1] * P[1][2] - P[0][2] * P[1][1]) * id;
    inv[1][0] = -(P[1][0] * P[2][2] - P[1][2] * P[2][0]) * id;
    inv[1][1] =  (P[0][0] * P[2][2] - P[0][2] * P[2][0]) * id;
    inv[1][2] = -(P[0][0] * P[1][2] - P[0][2] * P[1][0]) * id;
    inv[2][0] =  (P[1][0] * P[2][1] - P[1][1] * P[2][0]) * id;
    inv[2][1] = -(P[0][0] * P[2][1] - P[0][1] * P[2][0]) * id;
    inv[2][2] =  (P[0][0] * P[1][1] - P[0][1] * P[1][0]) * id;
    float rec[3][3];   // recip = inv(P)^T
    #pragma unroll
    for (int i = 0; i < 3; ++i)
        #pragma unroll
        for (int j = 0; j < 3; ++j)
            rec[i][j] = inv[j][i];

    float c0 = cosf(ang[0]), s0 = sinf(ang[0]);
    float c1 = cosf(ang[1]), s1 = sinf(ang[1]);
    float c2 = cosf(ang[2]), s2 = sinf(ang[2]);
    float rx[3][3] = {{1, 0, 0}, {0, c0, -s0}, {0, s0, c0}};
    float ry[3][3] = {{c1, 0, s1}, {0, 1, 0}, {-s1, 0, c1}};
    float rz[3][3] = {{c2, -s2, 0}, {s2, c2, 0}, {0, 0, 1}};
    float t[3][3], R[3][3], RL[3][3];
    mat3mul(t, ry, rz);
    mat3mul(R, rx, t);
    mat3mul(RL, R, rec);
    #pragma unroll
    for (int i = 0; i < 3; ++i)
        #pragma unroll
        for (int j = 0; j < 3; ++j)
            rl[3 * i + j] = RL[i][j];
}

// ---------------------------------------------------------------------------
// Kernel 1: pack exp rays into 16x16x4 A-fragment layout.
//   A is 16x4 (M x K): lane<16 holds (K0,K1)=(x,y) of row M=lane,
//                      lane>=16 holds (K2,K3)=(z,1.0) of row M=lane-16.
// ---------------------------------------------------------------------------
__global__ void prep_a_kernel(const float* __restrict__ uq_exp, float* __restrict__ apack,
                              int n, int n_pad) {
    int i = blockIdx.x * blockDim.x + threadIdx.x;
    if (i >= n_pad) return;
    float x = 0.0f, y = 0.0f, z = 0.0f;
    if (i < n) { x = uq_exp[3 * i]; y = uq_exp[3 * i + 1]; z = uq_exp[3 * i + 2]; }
    int tile = i >> 4, ml = i & 15;
    float* base = apack + (size_t)tile * 64;
    base[ml * 2 + 0] = x;            // lane=ml,    v0 = K0
    base[ml * 2 + 1] = y;            // lane=ml,    v1 = K1
    base[(ml + 16) * 2 + 0] = z;     // lane=ml+16, v0 = K2
    base[(ml + 16) * 2 + 1] = 1.0f;  // lane=ml+16, v1 = K3 (homogeneous 1 -> adds bias row)
}

// ---------------------------------------------------------------------------
// Kernel 2: theo rays -> unit q, energy filter, PAIRED B-fragment layout.
//   Tile pair p packs tiles (2p, 2p+1) so each lane reads one b128:
//     lane L, pair p:  bpack[p*128 + L*4 + {0,1}] = tile 2p   fragment elems
//                      bpack[p*128 + L*4 + {2,3}] = tile 2p+1 fragment elems
//   Within a tile: lane<16 holds (K0,K1)=(x,y) of col N=lane,
//                  lane>=16 holds (K2,K3)=(z,bias) of col N=lane-16.
//   bias = 0 (kept) or -2 (energy-masked / padding).
// ---------------------------------------------------------------------------
__global__ void prep_b_kernel(const int* __restrict__ hkl, const float* __restrict__ rl,
                              const float* __restrict__ e_min_p, float* __restrict__ bpack,
                              int m, int m_pad) {
    int j = blockIdx.x * blockDim.x + threadIdx.x;
    if (j >= m_pad) return;
    float x = 0.0f, y = 0.0f, z = 0.0f, bias = -2.0f;
    if (j < m) {
        float h = (float)hkl[3 * j], k = (float)hkl[3 * j + 1], l = (float)hkl[3 * j + 2];
        float qx = rl[0] * h + rl[1] * k + rl[2] * l;
        float qy = rl[3] * h + rl[4] * k + rl[5] * l;
        float qz = rl[6] * h + rl[7] * k + rl[8] * l;
        float qn = sqrtf(qx * qx + qy * qy + qz * qz);
        float inv = 1.0f / qn;
        float ux = qx * inv, uy = qy * inv, uz = qz * inv;
        float sin_t = fabsf(uz);
        float energy = HC_F * qn / (2.0f * fmaxf(sin_t, 1e-9f));
        if (energy >= e_min_p[0]) { x = ux; y = uy; z = uz; bias = 0.0f; }
    }
    int tile = j >> 4, ml = j & 15;
    int pair = tile >> 1, sub = tile & 1;     // which tile of the pair
    float* base = bpack + (size_t)pair * 128;
    base[ml * 4 + sub * 2 + 0] = x;           // lane=ml,    K0
    base[ml * 4 + sub * 2 + 1] = y;           // lane=ml,    K1
    base[(ml + 16) * 4 + sub * 2 + 0] = z;    // lane=ml+16, K2
    base[(ml + 16) * 4 + sub * 2 + 1] = bias; // lane=ml+16, K3 (bias row)
}

// ---------------------------------------------------------------------------
// Kernel 3 (main): per wave: FOUR 16-row exp tiles x one chunk of column tile
// PAIRS. Per pair: 1 global_load_b128, 8 independent v_wmma_f32_16x16x4_f32
// (16384 FLOP per load), then 32 v_max3_num_f32 fold both D fragments into
// the running row maxima (16 VALU per tile, no canonicalization).
// ---------------------------------------------------------------------------
__global__ void score_kernel(const float* __restrict__ apack, const float* __restrict__ bpack,
                             float* __restrict__ rowmax2d, int col_pairs, int num_chunks,
                             int n_pad) {
    int wave = threadIdx.x >> 5;
    int lane = threadIdx.x & 31;
    // readfirstlane -> compiler sees wave-uniform chunk: scalar loop control.
    int chunk = __builtin_amdgcn_readfirstlane(blockIdx.y * WAVES_PER_BLOCK + wave);
    if (chunk >= num_chunks) return;            // wave-uniform: EXEC stays all-1 for WMMA
    int rtb = blockIdx.x * ROW_TILES_PER_WAVE;  // first row tile of this wave's group

    v2f av[ROW_TILES_PER_WAVE];
    #pragma unroll
    for (int i = 0; i < ROW_TILES_PER_WAVE; ++i)
        av[i] = *(const v2f*)(apack + (size_t)(rtb + i) * 64 + lane * 2);

    float vmax[ROW_TILES_PER_WAVE][8];
    #pragma unroll
    for (int i = 0; i < ROW_TILES_PER_WAVE; ++i)
        #pragma unroll
        for (int r = 0; r < 8; ++r) vmax[i][r] = -4.0f;

    int p0 = chunk * CHUNK_PAIRS;
    int p1 = p0 + CHUNK_PAIRS;
    if (p1 > col_pairs) p1 = col_pairs;
    for (int p = p0; p < p1; ++p) {
        v4f b4 = *(const v4f*)(bpack + (size_t)p * 128 + lane * 4);
        v2f bv0; bv0.x = b4.x; bv0.y = b4.y;    // even tile of the pair
        v2f bv1; bv1.x = b4.z; bv1.y = b4.w;    // odd tile of the pair
        v8f d0[ROW_TILES_PER_WAVE], d1[ROW_TILES_PER_WAVE];
        #pragma unroll
        for (int i = 0; i < ROW_TILES_PER_WAVE; ++i) {
            v8f c = {};
            // D = A(16x4) x B(4x16) + 0 ; K3 row carries the energy-mask bias.
            d0[i] = __builtin_amdgcn_wmma_f32_16x16x4_f32(
                false, av[i], false, bv0, (short)0, c, false, false);
            d1[i] = __builtin_amdgcn_wmma_f32_16x16x4_f32(
                false, av[i], false, bv1, (short)0, c, false, false);
        }
        #pragma unroll
        for (int i = 0; i < ROW_TILES_PER_WAVE; ++i)
            #pragma unroll
            for (int r = 0; r < 8; ++r)
                vmax[i][r] = fmax3_raw(vmax[i][r], d0[i][r], d1[i][r]);
    }

    // Reduce across the 16 lanes (columns) sharing each row; xor masks 1..8
    // keep the reduction inside each 16-lane half of the wave32.
    #pragma unroll
    for (int i = 0; i < ROW_TILES_PER_WAVE; ++i) {
        #pragma unroll
        for (int r = 0; r < 8; ++r) {
            float v = vmax[i][r];
            v = fmax_raw(v, __shfl_xor(v, 1, 32));
            v = fmax_raw(v, __shfl_xor(v, 2, 32));
            v = fmax_raw(v, __shfl_xor(v, 4, 32));
            v = fmax_raw(v, __shfl_xor(v, 8, 32));
            vmax[i][r] = v;
        }
        // C/D layout: lanes 0-15 hold rows M=r, lanes 16-31 hold rows M=r+8.
        float* dst = rowmax2d + (size_t)chunk * n_pad + (size_t)(rtb + i) * 16;
        if (lane == 0) {
            #pragma unroll
            for (int r = 0; r < 8; ++r) dst[r] = vmax[i][r];
        } else if (lane == 16) {
            #pragma unroll
            for (int r = 0; r < 8; ++r) dst[8 + r] = vmax[i][r];
        }
    }
}

// ---------------------------------------------------------------------------
// Kernel 4: reduce chunk maxima per row, clip exactly like the reference,
// acos once per exp ray, Gaussian kernel, mean -> d_out[0].
// ---------------------------------------------------------------------------
__global__ void finalize_kernel(const float* __restrict__ rowmax2d,
                                const float* __restrict__ phi_max_p,
                                float* __restrict__ out, int n, int n_pad, int num_chunks) {
    __shared__ float sdata[256];
    int tid = threadIdx.x;
    float inv_pm = 1.0f / phi_max_p[0];
    float sum = 0.0f;
    for (int i = tid; i < n; i += blockDim.x) {
        float mx = -4.0f;
        for (int c = 0; c < num_chunks; ++c)
            mx = fmax_raw(mx, rowmax2d[(size_t)c * n_pad + i]);
        mx = fminf(fmaxf(mx, -1.0f + 1e-6f), 1.0f - 1e-6f);
        float phi = acosf(mx);
        float u = phi * inv_pm;
        sum += expf(-0.5f * u * u);
    }
    sdata[tid] = sum;
    __syncthreads();
    for (int s = 128; s > 0; s >>= 1) {
        if (tid < s) sdata[tid] += sdata[tid + s];
        __syncthreads();
    }
    if (tid == 0) out[0] = sdata[0] / (float)n;
}

// ---------------------------------------------------------------------------
extern "C" void kernel_launch(void* const* d_in, const int* in_sizes, int n_in,
                              void* d_out, int out_size, void* d_ws, size_t ws_size,
                              hipStream_t stream) {
    const float* lattice = (const float*)d_in[0];
    const float* angle   = (const float*)d_in[1];
    const float* uq_exp  = (const float*)d_in[2];
    const int*   hkl     = (const int*)d_in[3];
    const float* phi_max = (const float*)d_in[4];
    const float* e_min   = (const float*)d_in[5];
    float* out = (float*)d_out;

    int n = in_sizes[2] / 3;   // ~5000 experimental rays
    int m = in_sizes[3] / 3;   // ~12k theoretical rays
    int row_tiles = (n + 15) / 16;
    // pad row tiles so every wave owns a full group of ROW_TILES_PER_WAVE tiles
    int row_groups = (row_tiles + ROW_TILES_PER_WAVE - 1) / ROW_TILES_PER_WAVE;
    int row_tiles_pad = row_groups * ROW_TILES_PER_WAVE;
    int n_pad = row_tiles_pad * 16;
    // pad columns to a whole tile PAIR (32 rays)
    int col_pairs = (m + 31) / 32;
    int m_pad = col_pairs * 32;
    int num_chunks = (col_pairs + CHUNK_PAIRS - 1) / CHUNK_PAIRS;

    // Workspace layout (floats); everything is fully rewritten every call.
    float* ws       = (float*)d_ws;
    float* rl       = ws;                                  // 9 (pad to 16)
    float* apack    = ws + 16;                             // row_tiles_pad*64
    float* bpack    = apack + (size_t)row_tiles_pad * 64;  // col_pairs*128 (16B aligned)
    float* rowmax2d = bpack + (size_t)col_pairs * 128;     // num_chunks*n_pad

    prep_rl_kernel<<<1, 32, 0, stream>>>(lattice, angle, rl);
    prep_a_kernel<<<(n_pad + 255) / 256, 256, 0, stream>>>(uq_exp, apack, n, n_pad);
    prep_b_kernel<<<(m_pad + 255) / 256, 256, 0, stream>>>(hkl, rl, e_min, bpack, m, m_pad);

    dim3 grid(row_groups, (num_chunks + WAVES_PER_BLOCK - 1) / WAVES_PER_BLOCK);
    score_kernel<<<grid, WAVES_PER_BLOCK * 32, 0, stream>>>(
        apack, bpack, rowmax2d, col_pairs, num_chunks, n_pad);

    finalize_kernel<<<1, 256, 0, stream>>>(rowmax2d, phi_max, out, n, n_pad, num_chunks);
}